// TransformerLayer_29850022707312
// MI455X (gfx1250) — compile-verified
//
#include <hip/hip_runtime.h>
#include <math.h>

typedef float v2f __attribute__((ext_vector_type(2)));
typedef float v8f __attribute__((ext_vector_type(8)));

#define WMMA_F32(a, b, c) \
  __builtin_amdgcn_wmma_f32_16x16x4_f32(false, (a), false, (b), (short)0, (c), false, false)

// Problem constants (from setup_inputs): B=2, H=W=128, C=128, ns=4
constexpr int Bb   = 2;
constexpr int Hh   = 128;
constexpr int Ww   = 128;
constexpr int Cc   = 128;
constexpr int SHW  = 16;            // cyclic shift (h/ns)/2
constexpr int WA   = 1024;          // window area 32*32
constexpr int NWIN = Bb * 16;       // 32 windows
constexpr int NTOK = Bb * Hh * Ww;  // 32768 tokens
constexpr int HID  = 1024;

// Window-local element (win, l) -> merged token index, with roll applied.
// split_windows(roll(x,-s)) elem (win, yy, xx) == x[(wy*32+yy+s)%128, (wx*32+xx+s)%128]
__device__ __forceinline__ int win_row_to_token(int win, int l) {
  int b   = win >> 4;
  int wib = win & 15;
  int wy = wib >> 2, wx = wib & 3;
  int yy = l >> 5, xx = l & 31;
  int y = (wy * 32 + yy + SHW) & 127;
  int x = (wx * 32 + xx + SHW) & 127;
  return b * (Hh * Ww) + y * Ww + x;
}

// ---------------------------------------------------------------------------
// Kernel 1: q/k/v projections, written directly in windowed layout [32][1024][128]
// ---------------------------------------------------------------------------
__global__ __launch_bounds__(256) void k_qkv(
    const float* __restrict__ src, const float* __restrict__ tgt,
    const float* __restrict__ Wq, const float* __restrict__ Wk,
    const float* __restrict__ Wv,
    float* __restrict__ qw, float* __restrict__ kw, float* __restrict__ vw) {
  __shared__ float sS[16][132];
  __shared__ float sT[16][132];
  int win = blockIdx.x >> 6;   // / 64 row-tiles
  int rt  = blockIdx.x & 63;
  int tid = threadIdx.x;
  {
    int r = tid >> 4, ch = tid & 15;
    int tok = win_row_to_token(win, rt * 16 + r);
    const float* sp = src + (size_t)tok * Cc + ch * 8;
    const float* tp = tgt + (size_t)tok * Cc + ch * 8;
    *(float4*)&sS[r][ch * 8]     = *(const float4*)(sp);
    *(float4*)&sS[r][ch * 8 + 4] = *(const float4*)(sp + 4);
    *(float4*)&sT[r][ch * 8]     = *(const float4*)(tp);
    *(float4*)&sT[r][ch * 8 + 4] = *(const float4*)(tp + 4);
  }
  __syncthreads();

  int wv = tid >> 5, lane = tid & 31;
  int hi = lane >> 4, n = lane & 15, m = lane & 15;
  int colb = wv * 16;
  const float* wqp = Wq + (size_t)(colb + n) * Cc;
  const float* wkp = Wk + (size_t)(colb + n) * Cc;
  const float* wvp = Wv + (size_t)(colb + n) * Cc;
  v8f aq = {}, ak = {}, av = {};
#pragma unroll
  for (int kk = 0; kk < 32; ++kk) {
    int k0 = kk * 4 + 2 * hi;
    v2f as; as.x = sS[m][k0]; as.y = sS[m][k0 + 1];
    v2f at; at.x = sT[m][k0]; at.y = sT[m][k0 + 1];
    v2f bq; bq.x = wqp[k0]; bq.y = wqp[k0 + 1];
    v2f bk; bk.x = wkp[k0]; bk.y = wkp[k0 + 1];
    v2f bv; bv.x = wvp[k0]; bv.y = wvp[k0 + 1];
    aq = WMMA_F32(as, bq, aq);
    ak = WMMA_F32(at, bk, ak);
    av = WMMA_F32(at, bv, av);
  }
  size_t rowg = (size_t)win * WA + rt * 16;
#pragma unroll
  for (int rr = 0; rr < 8; ++rr) {
    int M = rr + 8 * hi;
    size_t off = (rowg + M) * Cc + colb + n;
    qw[off] = aq[rr]; kw[off] = ak[rr]; vw[off] = av[rr];
  }
}

// ---------------------------------------------------------------------------
// Kernel 2: flash attention per window; output stored merged+rolled [NTOK][C]
// ---------------------------------------------------------------------------
__global__ __launch_bounds__(256) void k_attn(
    const float* __restrict__ qw, const float* __restrict__ kw,
    const float* __restrict__ vw, const float* __restrict__ mask,
    float* __restrict__ om) {
  __shared__ float sP[8][16][18];  // per-wave P staging (D-layout -> A-layout)
  int win  = blockIdx.x >> 3;
  int tid  = threadIdx.x;
  int wv   = tid >> 5, lane = tid & 31;
  int hi   = lane >> 4, n = lane & 15;
  int qt   = (blockIdx.x & 7) * 8 + wv;  // query tile 0..63

  const float* qrow = qw + ((size_t)win * WA + qt * 16 + n) * Cc;
  v2f qf[32];
#pragma unroll
  for (int kk = 0; kk < 32; ++kk) {
    int k0 = kk * 4 + 2 * hi;
    qf[kk].x = qrow[k0]; qf[kk].y = qrow[k0 + 1];
  }

  float rm[8], rl[8];
#pragma unroll
  for (int rr = 0; rr < 8; ++rr) { rm[rr] = -1e30f; rl[rr] = 0.f; }
  v8f zero = {};
  v8f acc[8];
#pragma unroll
  for (int t = 0; t < 8; ++t) acc[t] = zero;

  const float* mwin = mask + (size_t)(win & 15) * WA * WA;
  const float inv_scale = 0.088388347648318447f;  // 1/sqrt(128)

  for (int kt = 0; kt < 64; ++kt) {
    // S tile = Q * K^T
    const float* kbase = kw + ((size_t)win * WA + kt * 16 + n) * Cc;
    v8f s = zero;
#pragma unroll
    for (int kk = 0; kk < 32; ++kk) {
      int k0 = kk * 4 + 2 * hi;
      v2f bk; bk.x = kbase[k0]; bk.y = kbase[k0 + 1];
      s = WMMA_F32(qf[kk], bk, s);
    }
    // scale + mask + online softmax (row reductions across 16-lane halves)
    float pv[8], al[8];
#pragma unroll
    for (int rr = 0; rr < 8; ++rr) {
      int qr = qt * 16 + rr + 8 * hi;
      float svv = s[rr] * inv_scale + mwin[(size_t)qr * WA + kt * 16 + n];
      float tm = svv;
      tm = fmaxf(tm, __shfl_xor(tm, 1, 32));
      tm = fmaxf(tm, __shfl_xor(tm, 2, 32));
      tm = fmaxf(tm, __shfl_xor(tm, 4, 32));
      tm = fmaxf(tm, __shfl_xor(tm, 8, 32));
      float mnew = fmaxf(rm[rr], tm);
      al[rr] = __expf(rm[rr] - mnew);
      rm[rr] = mnew;
      float p = __expf(svv - mnew);
      pv[rr] = p;
      float ts = p;
      ts += __shfl_xor(ts, 1, 32);
      ts += __shfl_xor(ts, 2, 32);
      ts += __shfl_xor(ts, 4, 32);
      ts += __shfl_xor(ts, 8, 32);
      rl[rr] = rl[rr] * al[rr] + ts;
    }
#pragma unroll
    for (int t = 0; t < 8; ++t)
#pragma unroll
      for (int rr = 0; rr < 8; ++rr) acc[t][rr] *= al[rr];

    // stage P through LDS: D-layout store, A-layout load (cross-lane transpose)
    __syncthreads();
#pragma unroll
    for (int rr = 0; rr < 8; ++rr) sP[wv][rr + 8 * hi][n] = pv[rr];
    __syncthreads();
    v2f pa[4];
#pragma unroll
    for (int k2 = 0; k2 < 4; ++k2) {
      int k0 = k2 * 4 + 2 * hi;
      pa[k2].x = sP[wv][n][k0];
      pa[k2].y = sP[wv][n][k0 + 1];
    }
    // acc += P * V
    const float* vbase = vw + ((size_t)win * WA + kt * 16) * Cc;
#pragma unroll
    for (int t = 0; t < 8; ++t) {
      int d = t * 16 + n;
#pragma unroll
      for (int k2 = 0; k2 < 4; ++k2) {
        int s0 = k2 * 4 + 2 * hi;
        v2f bv; bv.x = vbase[(size_t)s0 * Cc + d]; bv.y = vbase[(size_t)(s0 + 1) * Cc + d];
        acc[t] = WMMA_F32(pa[k2], bv, acc[t]);
      }
    }
  }

  // normalize and scatter to merged+rolled layout
#pragma unroll
  for (int rr = 0; rr < 8; ++rr) {
    int l   = qt * 16 + rr + 8 * hi;
    int tok = win_row_to_token(win, l);
    float inv = 1.0f / rl[rr];
#pragma unroll
    for (int t = 0; t < 8; ++t)
      om[(size_t)tok * Cc + t * 16 + n] = acc[t][rr] * inv;
  }
}

// ---------------------------------------------------------------------------
// Kernel 3: msg = LN(attn_out @ Wm.T) * g1 + b1
// ---------------------------------------------------------------------------
__global__ __launch_bounds__(256) void k_msg_ln(
    const float* __restrict__ xin, const float* __restrict__ Wm,
    const float* __restrict__ g1, const float* __restrict__ b1,
    float* __restrict__ msg) {
  __shared__ float sA[16][132];
  __shared__ float sO[16][128];
  __shared__ float sMean[16], sRstd[16];
  int rowb = blockIdx.x * 16;
  int tid = threadIdx.x;
  {
    int r = tid >> 4, ch = tid & 15;
    const float* p = xin + (size_t)(rowb + r) * Cc + ch * 8;
    *(float4*)&sA[r][ch * 8]     = *(const float4*)p;
    *(float4*)&sA[r][ch * 8 + 4] = *(const float4*)(p + 4);
  }
  __syncthreads();
  int wv = tid >> 5, lane = tid & 31, hi = lane >> 4, n = lane & 15, m = lane & 15;
  int colb = wv * 16;
  const float* wmp = Wm + (size_t)(colb + n) * Cc;
  v8f a = {};
#pragma unroll
  for (int kk = 0; kk < 32; ++kk) {
    int k0 = kk * 4 + 2 * hi;
    v2f av; av.x = sA[m][k0]; av.y = sA[m][k0 + 1];
    v2f bv; bv.x = wmp[k0]; bv.y = wmp[k0 + 1];
    a = WMMA_F32(av, bv, a);
  }
#pragma unroll
  for (int rr = 0; rr < 8; ++rr) sO[rr + 8 * hi][colb + n] = a[rr];
  __syncthreads();
  if (tid < 16) {
    float s = 0.f, s2 = 0.f;
    for (int c = 0; c < 128; ++c) { float x = sO[tid][c]; s += x; s2 += x * x; }
    float mean = s * (1.f / 128.f);
    float var  = s2 * (1.f / 128.f) - mean * mean;
    sMean[tid] = mean;
    sRstd[tid] = rsqrtf(var + 1e-5f);
  }
  __syncthreads();
  {
    int r = tid >> 4, ch = (tid & 15) * 8;
    float mean = sMean[r], rstd = sRstd[r];
    float* op = msg + (size_t)(rowb + r) * Cc;
#pragma unroll
    for (int i = 0; i < 8; ++i) {
      int c = ch + i;
      op[c] = (sO[r][c] - mean) * rstd * g1[c] + b1[c];
    }
  }
}

// ---------------------------------------------------------------------------
// Kernel 4: fused FFN: out = src + LN2( gelu([src,msg] @ W1.T) @ W2.T )
// ---------------------------------------------------------------------------
__global__ __launch_bounds__(256) void k_ffn(
    const float* __restrict__ src, const float* __restrict__ msg,
    const float* __restrict__ W1, const float* __restrict__ W2,
    const float* __restrict__ g2, const float* __restrict__ b2,
    float* __restrict__ out) {
  __shared__ float sC[16][264];   // cat tile [16][256] (+pad)
  __shared__ float sH[16][132];   // hidden chunk / final msg2 tile
  __shared__ float sMean[16], sRstd[16];
  int rowb = blockIdx.x * 16;
  int tid = threadIdx.x;
  {
    int r = tid >> 4, seg = tid & 15;
    const float* p = (seg < 8) ? (src + (size_t)(rowb + r) * Cc + seg * 16)
                               : (msg + (size_t)(rowb + r) * Cc + (seg - 8) * 16);
    *(float4*)&sC[r][seg * 16]      = *(const float4*)(p);
    *(float4*)&sC[r][seg * 16 + 4]  = *(const float4*)(p + 4);
    *(float4*)&sC[r][seg * 16 + 8]  = *(const float4*)(p + 8);
    *(float4*)&sC[r][seg * 16 + 12] = *(const float4*)(p + 12);
  }
  __syncthreads();

  int wv = tid >> 5, lane = tid & 31, hi = lane >> 4, n = lane & 15, m = lane & 15;
  v8f ao = {};
  for (int chv = 0; chv < 8; ++chv) {
    // stage 1: hidden chunk cols [chv*128 .. +127], this wave does 16 of them
    int j = chv * 128 + wv * 16 + n;
    const float* w1p = W1 + (size_t)j * 256;
    v8f ah = {};
#pragma unroll
    for (int kk = 0; kk < 64; ++kk) {
      int k0 = kk * 4 + 2 * hi;
      v2f av; av.x = sC[m][k0]; av.y = sC[m][k0 + 1];
      v2f bv; bv.x = w1p[k0]; bv.y = w1p[k0 + 1];
      ah = WMMA_F32(av, bv, ah);
    }
    __syncthreads();  // protect sH from previous chunk's stage-2 reads
#pragma unroll
    for (int rr = 0; rr < 8; ++rr) {
      float x = ah[rr];
      sH[rr + 8 * hi][wv * 16 + n] = 0.5f * x * (1.f + erff(x * 0.70710678118654752f));
    }
    __syncthreads();
    // stage 2: out cols [wv*16 .. +15] accumulate over this hidden chunk
    const float* w2p = W2 + (size_t)(wv * 16 + n) * HID + chv * 128;
#pragma unroll
    for (int kk = 0; kk < 32; ++kk) {
      int k0 = kk * 4 + 2 * hi;
      v2f av; av.x = sH[m][k0]; av.y = sH[m][k0 + 1];
      v2f bv; bv.x = w2p[k0]; bv.y = w2p[k0 + 1];
      ao = WMMA_F32(av, bv, ao);
    }
  }
  __syncthreads();
#pragma unroll
  for (int rr = 0; rr < 8; ++rr) sH[rr + 8 * hi][wv * 16 + n] = ao[rr];
  __syncthreads();
  if (tid < 16) {
    float s = 0.f, s2 = 0.f;
    for (int c = 0; c < 128; ++c) { float x = sH[tid][c]; s += x; s2 += x * x; }
    float mean = s * (1.f / 128.f);
    float var  = s2 * (1.f / 128.f) - mean * mean;
    sMean[tid] = mean;
    sRstd[tid] = rsqrtf(var + 1e-5f);
  }
  __syncthreads();
  {
    int r = tid >> 4, ch = (tid & 15) * 8;
    float mean = sMean[r], rstd = sRstd[r];
    const float* sp = src + (size_t)(rowb + r) * Cc;
    float* op = out + (size_t)(rowb + r) * Cc;
#pragma unroll
    for (int i = 0; i < 8; ++i) {
      int c = ch + i;
      op[c] = sp[c] + (sH[r][c] - mean) * rstd * g2[c] + b2[c];
    }
  }
}

// ---------------------------------------------------------------------------
extern "C" void kernel_launch(void* const* d_in, const int* in_sizes, int n_in,
                              void* d_out, int out_size, void* d_ws, size_t ws_size,
                              hipStream_t stream) {
  const float* src  = (const float*)d_in[0];
  const float* tgt  = (const float*)d_in[1];
  const float* mask = (const float*)d_in[2];
  const float* Wq   = (const float*)d_in[3];
  const float* Wk   = (const float*)d_in[4];
  const float* Wv   = (const float*)d_in[5];
  const float* Wm   = (const float*)d_in[6];
  const float* g1   = (const float*)d_in[7];
  const float* b1   = (const float*)d_in[8];
  const float* W1   = (const float*)d_in[9];
  const float* W2   = (const float*)d_in[10];
  const float* g2   = (const float*)d_in[11];
  const float* b2   = (const float*)d_in[12];
  float* out = (float*)d_out;

  float* ws = (float*)d_ws;
  const size_t PWIN = (size_t)NWIN * WA * Cc;  // 4,194,304 floats
  float* qw = ws;
  float* kw = qw + PWIN;
  float* vw = kw + PWIN;
  float* om = vw + PWIN;                 // attn output, merged layout [NTOK][C]
  float* ms = om + (size_t)NTOK * Cc;    // LN'd message

  k_qkv   <<<dim3(NWIN * 64), dim3(256), 0, stream>>>(src, tgt, Wq, Wk, Wv, qw, kw, vw);
  k_attn  <<<dim3(NWIN * 8),  dim3(256), 0, stream>>>(qw, kw, vw, mask, om);
  k_msg_ln<<<dim3(NTOK / 16), dim3(256), 0, stream>>>(om, Wm, g1, b1, ms);
  k_ffn   <<<dim3(NTOK / 16), dim3(256), 0, stream>>>(src, ms, W1, W2, g2, b2, out);
}